// NSLayer_50062138802796
// MI455X (gfx1250) — compile-verified
//
#include <hip/hip_runtime.h>

// CDNA5 / gfx1250: fp32 WMMA path, wave32.
typedef __attribute__((ext_vector_type(2))) float v2f;  // A/B frag for 16x16x4 f32 (2 VGPRs)
typedef __attribute__((ext_vector_type(4))) float v4f;  // b128 LDS moves
typedef __attribute__((ext_vector_type(8))) float v8f;  // C/D 16x16 f32 (8 VGPRs)

#define WAVES_PER_BLOCK 8
// 20 floats: row starts 16B-aligned (20r+8h == 0 mod 4 words) and the 16 rows
// map to 16 distinct banks (step 20 mod 64 has a 16-long cycle).
#define LDS_STRIDE 20

// One wave handles one 16x16 matrix:
//   A = I - X*X^T ; six squarings ; Mat = sum w[i]*M_i ; out = X + Mat*X
// All round-tripped matrices (S, A, powers, Mat) are bitwise symmetric, so:
//   * B-frag == A-frag for every squaring (no column loads)
//   * C-layout register block stores to LDS as contiguous rows (2x ds_store_b128)
__global__ __launch_bounds__(WAVES_PER_BLOCK * 32)
void ns_layer_poly_kernel(const float* __restrict__ in,
                          const float* __restrict__ w7,
                          float* __restrict__ out,
                          int nmat)
{
    const int wave = threadIdx.x >> 5;
    const int lane = threadIdx.x & 31;
    const int mid  = blockIdx.x * WAVES_PER_BLOCK + wave;   // matrix id (wave-uniform)
    if (mid >= nmat) return;   // wave-uniform: EXEC stays all-ones for WMMA

    __shared__ float ldsM[WAVES_PER_BLOCK][16 * LDS_STRIDE]; // current symmetric power
    __shared__ float ldsX[WAVES_PER_BLOCK][16 * LDS_STRIDE]; // X^T stash
    float* Mt = ldsM[wave];
    float* Xt = ldsX[wave];

    const int r = lane & 15;   // row index (A-frag) / N (C-layout)
    const int h = lane >> 4;   // lane-half selects K/M sub-block

    const float* src = in + (size_t)mid * 256;

    // Broadcast weights (uniform -> scalar loads)
    float w[7];
#pragma unroll
    for (int i = 0; i < 7; ++i) w[i] = w7[i];

    // ---- Load X directly in fragment layout: xfrag[k] = (X[r][4k+2h], X[r][4k+2h+1])
    v2f xfrag[4];
#pragma unroll
    for (int k = 0; k < 4; ++k) {
        const float* p = src + r * 16 + 4 * k + 2 * h;      // 8B aligned
        xfrag[k] = *(const v2f*)p;                          // global_load_b64
    }
    // Stash X TRANSPOSED: Xt[c][r] = X[r][c]  (turns later column reads into row reads)
#pragma unroll
    for (int k = 0; k < 4; ++k) {
        const int c = 4 * k + 2 * h;
        Xt[c * LDS_STRIDE + r]       = xfrag[k].x;
        Xt[(c + 1) * LDS_STRIDE + r] = xfrag[k].y;
    }

    // ---- S = X * X^T : for this symmetric product, B-frag == A-frag == xfrag
    v8f s = {0.f, 0.f, 0.f, 0.f, 0.f, 0.f, 0.f, 0.f};
#pragma unroll
    for (int k = 0; k < 4; ++k) {
        s = __builtin_amdgcn_wmma_f32_16x16x4_f32(
                /*neg_a=*/false, xfrag[k], /*neg_b=*/false, xfrag[k],
                /*c_mod=*/(short)0, s, /*reuse_a=*/false, /*reuse_b=*/false);
    }

    // ---- A = I - S   (C-layout: element (M=v+8h, N=r); diagonal where v+8h==r)
    v8f cur;
#pragma unroll
    for (int v = 0; v < 8; ++v)
        cur[v] = ((v + 8 * h) == r ? 1.0f : 0.0f) - s[v];

    // ---- acc = w0 * A
    v8f acc;
#pragma unroll
    for (int v = 0; v < 8; ++v) acc[v] = w[0] * cur[v];

    // ---- Six repeated squarings through the wave-private LDS tile
#pragma unroll
    for (int it = 1; it <= 6; ++it) {
        // cur is symmetric: lane's C-layout block (col r, rows 8h..8h+7) equals
        // row r, cols 8h..8h+7  ->  two contiguous 16B stores.
        *(v4f*)&Mt[r * LDS_STRIDE + 8 * h]     = __builtin_shufflevector(cur, cur, 0, 1, 2, 3);
        *(v4f*)&Mt[r * LDS_STRIDE + 8 * h + 4] = __builtin_shufflevector(cur, cur, 4, 5, 6, 7);
        // no barrier needed: tile is wave-private; DS ops are in-order per wave

        v8f nd = {0.f, 0.f, 0.f, 0.f, 0.f, 0.f, 0.f, 0.f};
#pragma unroll
        for (int k = 0; k < 4; ++k) {
            // symmetric operand: A-frag and B-frag are the same row pair
            v2f a = *(const v2f*)&Mt[r * LDS_STRIDE + 4 * k + 2 * h];   // ds_load_b64
            nd = __builtin_amdgcn_wmma_f32_16x16x4_f32(
                     false, a, false, a, (short)0, nd, false, false);
        }
        cur = nd;
#pragma unroll
        for (int v = 0; v < 8; ++v) acc[v] += w[it] * cur[v];
    }

    // ---- out = X + Mat*X : seed D with X (C-layout) and accumulate Mat*X on top
    // Mat (acc) is symmetric -> same contiguous row store.
    *(v4f*)&Mt[r * LDS_STRIDE + 8 * h]     = __builtin_shufflevector(acc, acc, 0, 1, 2, 3);
    *(v4f*)&Mt[r * LDS_STRIDE + 8 * h + 4] = __builtin_shufflevector(acc, acc, 4, 5, 6, 7);

    // D := X in C-layout: X[v+8h][r] == Xt[r][v+8h] -> two contiguous 16B loads
    v8f dout;
    {
        v4f lo = *(const v4f*)&Xt[r * LDS_STRIDE + 8 * h];
        v4f hi = *(const v4f*)&Xt[r * LDS_STRIDE + 8 * h + 4];
#pragma unroll
        for (int v = 0; v < 4; ++v) { dout[v] = lo[v]; dout[v + 4] = hi[v]; }
    }

#pragma unroll
    for (int k = 0; k < 4; ++k) {
        const int kb = 4 * k + 2 * h;
        v2f a = *(const v2f*)&Mt[r * LDS_STRIDE + kb];      // Mat row pair (symmetric)
        v2f b = *(const v2f*)&Xt[r * LDS_STRIDE + kb];      // B=X: (X[kb][r],X[kb+1][r]) = Xt row pair
        dout = __builtin_amdgcn_wmma_f32_16x16x4_f32(
                   false, a, false, b, (short)0, dout, false, false);
    }

    // ---- Store result (C-layout: each b32 store covers two 64B segments across the wave)
    float* dst = out + (size_t)mid * 256;
#pragma unroll
    for (int v = 0; v < 8; ++v)
        dst[(v + 8 * h) * 16 + r] = dout[v];
}

extern "C" void kernel_launch(void* const* d_in, const int* in_sizes, int n_in,
                              void* d_out, int out_size, void* d_ws, size_t ws_size,
                              hipStream_t stream) {
    const float* inp = (const float*)d_in[0];   // [4096,64,16,16] f32
    const float* w   = (const float*)d_in[1];   // 7 f32 weights
    float* outp      = (float*)d_out;

    const int nmat   = in_sizes[0] / 256;       // 262144 matrices
    const int blocks = (nmat + WAVES_PER_BLOCK - 1) / WAVES_PER_BLOCK;

    ns_layer_poly_kernel<<<blocks, WAVES_PER_BLOCK * 32, 0, stream>>>(inp, w, outp, nmat);
}